// PoolingLayerUnet_38534446579901
// MI455X (gfx1250) — compile-verified
//
#include <hip/hip_runtime.h>

// MI455X (gfx1250) implementation.
// Roofline: one 1.024 GB streaming read of x -> ~44us @ 23.3 TB/s; everything
// else (y = 8 MB, gather = 1 MB) is L2-resident. Matvec keeps f32 precision
// (top-k threshold sensitivity rules out reduced-precision WMMA) and uses
// V_WMMA_F32_16X16X4_F32 with v in the A slot (row M=0) so the 16 dots land
// in acc[0] of lanes 0..15 -> coalesced y store.
//
// CDNA5 data movement: tiles are staged memory->LDS with
// GLOBAL_LOAD_ASYNC_TO_LDS_B128 (ASYNCcnt), double-buffered per wave:
// issue next tile's 16 async loads, s_wait_asynccnt 0x10 (in-order done ->
// current tile resident), run 32 WMMAs. LDS layout: 512B rows with a 16B-chunk
// XOR swizzle (chunk c of row r stored at c^r) -> conflict-free b64 fragment
// reads across all 64 banks without padding (async op shares one offset for
// both LDS and global sides, so padded strides are not usable).
//
// Workspace layout (needs 4096 + 4*N + 4*k bytes ~= 8.01 MB):
//   u32[0] scale(bits)  u32[1] selCnt  u32[2] eqCnt  u32[3] thresh
//   u32[4] needEq       u32[5] prefix  u32[6] remaining-k
//   u32[16..272) 256-bin histogram
//   byte 4096: y[N] f32;  byte 4096+4N: idx list [k] u32

typedef __attribute__((ext_vector_type(2))) float v2f;
typedef __attribute__((ext_vector_type(8))) float v8f;

#define D 128
#define HDR_SCALE  0
#define HDR_SELCNT 1
#define HDR_EQCNT  2
#define HDR_THRESH 3
#define HDR_NEEDEQ 4
#define HDR_PREFIX 5
#define HDR_REMAIN 6
#define HIST_OFF   16
#define WPB        2            // waves per block (wave32)
#define TILE_F     2048         // floats per 16x128 tile buffer
#define WAVE_F     (2 * TILE_F) // double buffer per wave

// Monotonic float -> uint map (descending float == descending uint)
__device__ __forceinline__ unsigned mapf(float f) {
  unsigned u = __float_as_uint(f);
  return (u & 0x80000000u) ? ~u : (u | 0x80000000u);
}

__global__ __launch_bounds__(256)
void k_init(const float* __restrict__ v, const int* __restrict__ kp,
            unsigned* __restrict__ hdr, float* __restrict__ out, int N) {
  __shared__ float red[128];
  int t = threadIdx.x;
  if (t < 128) { float a = v[t]; red[t] = a * a; }
  __syncthreads();
  for (int s = 64; s > 0; s >>= 1) {
    if (t < s) red[t] += red[t + s];
    __syncthreads();
  }
  if (t == 0) {
    ((float*)hdr)[HDR_SCALE] = 1.0f / sqrtf(red[0]);
    int k = *kp; if (k > N) k = N;
    hdr[HDR_SELCNT] = 0; hdr[HDR_EQCNT] = 0;
    hdr[HDR_THRESH] = 0; hdr[HDR_NEEDEQ] = 0;
    hdr[HDR_PREFIX] = 0; hdr[HDR_REMAIN] = (unsigned)k;
  }
  if (t < 256) hdr[HIST_OFF + t] = 0;
  if (t < 128) out[t] = 0.0f;
}

// Issue 16 async B128 loads: memory row r -> LDS row r with 16B-chunk XOR swizzle.
__device__ __forceinline__ void stage_async(unsigned ldsBufByte,
                                            unsigned long long gtile, int lane) {
#pragma unroll
  for (int r = 0; r < 16; ++r) {
    unsigned ldsoff = ldsBufByte + (unsigned)(r * 512) + ((unsigned)(lane ^ r) << 4);
    unsigned long long ga = gtile + (unsigned)(r * 512);
    asm volatile("global_load_async_to_lds_b128 %0, %1, off"
                 :: "v"(ldsoff), "v"(ga) : "memory");
  }
}

// Guarded fallback for tail tiles (same swizzled layout, via VGPRs + ds_store).
__device__ __forceinline__ void stage_sync(float* buf, const float* x,
                                           long long r0, int N, int lane) {
#pragma unroll
  for (int r = 0; r < 16; ++r) {
    long long row = r0 + r;
    float4 g = make_float4(0.f, 0.f, 0.f, 0.f);
    if (row < N)
      g = *reinterpret_cast<const float4*>(&x[row * D + 4 * lane]);
    *reinterpret_cast<float4*>(&buf[r * D + ((lane ^ r) << 2)]) = g;
  }
}

// y[i] = dot(x[i,:], v) * scale via V_WMMA_F32_16X16X4_F32, async double-buffered.
__global__ __launch_bounds__(64)
void k_matvec(const float* __restrict__ x, const float* __restrict__ v,
              const unsigned* __restrict__ hdr, float* __restrict__ y, int N) {
  __shared__ float lds_v[D];
  __shared__ float lds_x[WPB * WAVE_F];   // 2 waves * 2 buffers * 16x128 = 32 KB
  const int t    = threadIdx.x;
  const int lane = t & 31;
  const int wave = t >> 5;
  if (t < D) lds_v[t] = v[t];
  __syncthreads();

  const float scale = ((const float*)hdr)[HDR_SCALE];
  const int   row16 = lane & 15;
  const int   koff  = (lane >> 4) << 1;           // lanes>=16 take K=2,3
  const float msel  = (row16 == 0) ? 1.0f : 0.0f; // A: only row M=0 carries v

  // A fragments (tile-invariant): lane0 -> v[4c],v[4c+1]; lane16 -> v[4c+2],v[4c+3]
  v2f afrag[32];
#pragma unroll
  for (int c = 0; c < 32; ++c) {
    afrag[c].x = msel * lds_v[4 * c + koff];
    afrag[c].y = msel * lds_v[4 * c + koff + 1];
  }

  const unsigned myByte = (unsigned)(size_t)(void*)&lds_x[0] +
                          (unsigned)(wave * WAVE_F * 4);
  float* myBuf = &lds_x[wave * WAVE_F];
  const unsigned long long gx = (unsigned long long)(size_t)(const void*)x +
                                (unsigned)(lane * 16);

  const int nTiles = (N + 15) >> 4;
  const int wgid   = blockIdx.x * WPB + wave;
  const int nWaves = gridDim.x * WPB;

  // Prologue: stage first tile into buffer 0
  if (wgid < nTiles) {
    long long r0 = (long long)wgid * 16;
    if (r0 + 16 <= N) stage_async(myByte, gx + (unsigned long long)r0 * 512ull, lane);
    else              stage_sync(myBuf, x, r0, N, lane);
  }

  unsigned bufSel = 0;
  for (int tile = wgid; tile < nTiles; tile += nWaves) {
    const long long r0 = (long long)tile * 16;
    // Prefetch next tile into the other buffer
    int next = tile + nWaves;
    bool nextAsync = false;
    if (next < nTiles) {
      long long nr0 = (long long)next * 16;
      unsigned nb = bufSel ^ 1u;
      if (nr0 + 16 <= N) {
        stage_async(myByte + nb * (TILE_F * 4u),
                    gx + (unsigned long long)nr0 * 512ull, lane);
        nextAsync = true;
      } else {
        stage_sync(&myBuf[nb * TILE_F], x, nr0, N, lane);
      }
    }
    // In-order ASYNCcnt: <=16 outstanding means current tile is resident.
    if (nextAsync) asm volatile("s_wait_asynccnt 0x10" ::: "memory");
    else           asm volatile("s_wait_asynccnt 0x0"  ::: "memory");

    const float* bp = &myBuf[bufSel * TILE_F];
    v8f acc = {};
#pragma unroll
    for (int c = 0; c < 32; ++c) {
      // chunk c of row row16 lives at swizzled chunk c^row16 (conflict-free b64)
      v2f bf = *reinterpret_cast<const v2f*>(
          &bp[row16 * D + ((c ^ row16) << 2) + koff]);
      acc = __builtin_amdgcn_wmma_f32_16x16x4_f32(
          false, afrag[c], false, bf, (short)0, acc, false, false);
    }
    // D row M=0 = acc[0] of lanes 0..15 -> coalesced store of 16 y's
    if (lane < 16) {
      long long row = r0 + lane;
      if (row < N) y[row] = acc[0] * scale;
    }
    bufSel ^= 1u;
  }
}

__global__ __launch_bounds__(256)
void k_hist(const float* __restrict__ y, int N, unsigned* __restrict__ hdr,
            unsigned prefMask, int shift) {
  __shared__ unsigned lh[256];
  unsigned* hist = hdr + HIST_OFF;
  int t = threadIdx.x;
  lh[t] = 0;
  __syncthreads();
  const unsigned prefix = hdr[HDR_PREFIX];
  int stride = gridDim.x * blockDim.x;
  for (int i = blockIdx.x * blockDim.x + t; i < N; i += stride) {
    unsigned u = mapf(y[i]);
    if (((u ^ prefix) & prefMask) == 0)
      atomicAdd(&lh[(u >> shift) & 255u], 1u);
  }
  __syncthreads();
  if (lh[t]) atomicAdd(&hist[t], lh[t]);
}

__global__ __launch_bounds__(256)
void k_scan(unsigned* __restrict__ hdr, int shift, int last) {
  unsigned* hist = hdr + HIST_OFF;
  if (threadIdx.x == 0) {
    unsigned r = hdr[HDR_REMAIN];
    unsigned g = 0, chosen = 0;
    for (int d = 255; d >= 0; --d) {
      unsigned c = hist[d];
      if (g + c >= r) { chosen = (unsigned)d; break; }
      g += c;
    }
    hdr[HDR_PREFIX] |= chosen << shift;
    hdr[HDR_REMAIN] = r - g;
    if (last) {
      hdr[HDR_THRESH] = hdr[HDR_PREFIX];
      hdr[HDR_NEEDEQ] = r - g;
    }
  }
  __syncthreads();
  for (int i = threadIdx.x; i < 256; i += blockDim.x) hist[i] = 0;
}

__global__ __launch_bounds__(256)
void k_select(const float* __restrict__ y, int N, unsigned* __restrict__ hdr,
              unsigned* __restrict__ idxl) {
  const unsigned T = hdr[HDR_THRESH];
  const unsigned needEq = hdr[HDR_NEEDEQ];
  int stride = gridDim.x * blockDim.x;
  for (int i = blockIdx.x * blockDim.x + threadIdx.x; i < N; i += stride) {
    unsigned u = mapf(y[i]);
    if (u > T) {
      unsigned p = atomicAdd(&hdr[HDR_SELCNT], 1u);
      idxl[p] = (unsigned)i;
    } else if (u == T) {
      unsigned e = atomicAdd(&hdr[HDR_EQCNT], 1u);
      if (e < needEq) {
        unsigned p = atomicAdd(&hdr[HDR_SELCNT], 1u);
        idxl[p] = (unsigned)i;
      }
    }
  }
}

// out[t] = sum_j sigmoid(y[idx_j]) * x[idx_j, t]  (1 MB gather, L2-resident)
__global__ __launch_bounds__(128)
void k_final(const float* __restrict__ x, const float* __restrict__ y,
             const unsigned* __restrict__ idxl, const int* __restrict__ kp,
             int N, float* __restrict__ out) {
  int k = *kp; if (k > N) k = N;
  const int t = threadIdx.x;   // column 0..127
  float acc = 0.0f;
  for (int j = blockIdx.x; j < k; j += gridDim.x) {
    unsigned id = idxl[j];
    float g = 1.0f / (1.0f + expf(-y[id]));
    acc += g * x[(long long)id * D + t];
  }
  atomicAdd(&out[t], acc);
}

extern "C" void kernel_launch(void* const* d_in, const int* in_sizes, int n_in,
                              void* d_out, int out_size, void* d_ws, size_t ws_size,
                              hipStream_t stream) {
  const float* x  = (const float*)d_in[0];
  const float* v  = (const float*)d_in[1];
  const int*   kp = (const int*)d_in[2];
  float* out = (float*)d_out;
  const int N = in_sizes[0] / D;

  unsigned* hdr  = (unsigned*)d_ws;
  float*    y    = (float*)((char*)d_ws + 4096);
  unsigned* idxl = (unsigned*)((char*)d_ws + 4096 + (size_t)N * 4);

  k_init<<<1, 256, 0, stream>>>(v, kp, hdr, out, N);
  k_matvec<<<4096, 64, 0, stream>>>(x, v, hdr, y, N);   // 8192 waves, ~15 tiles each
  // exact top-k threshold: 4x8-bit MSB radix select on mapped keys (y is L2-hot)
  k_hist<<<1024, 256, 0, stream>>>(y, N, hdr, 0x00000000u, 24);
  k_scan<<<1, 256, 0, stream>>>(hdr, 24, 0);
  k_hist<<<1024, 256, 0, stream>>>(y, N, hdr, 0xFF000000u, 16);
  k_scan<<<1, 256, 0, stream>>>(hdr, 16, 0);
  k_hist<<<1024, 256, 0, stream>>>(y, N, hdr, 0xFFFF0000u, 8);
  k_scan<<<1, 256, 0, stream>>>(hdr, 8, 0);
  k_hist<<<1024, 256, 0, stream>>>(y, N, hdr, 0xFFFFFF00u, 0);
  k_scan<<<1, 256, 0, stream>>>(hdr, 0, 1);
  k_select<<<1024, 256, 0, stream>>>(y, N, hdr, idxl);
  k_final<<<16, 128, 0, stream>>>(x, y, idxl, kp, N, out);
}